// OptimalTransport_59639915872741
// MI455X (gfx1250) — compile-verified
//
#include <hip/hip_runtime.h>
#include <hip/hip_bf16.h>

typedef __attribute__((ext_vector_type(2))) float v2f;
typedef __attribute__((ext_vector_type(8))) float v8f;

#define SEQ 1024       // s == t == 1024
#define DIM 768
#define TILE 64        // output tile per block
#define BK 32          // K-slab staged in LDS (8 WMMA k-steps)
#define LDSP 36        // padded LDS row stride (floats): %4==0 (16B chunks), 36*l mod 64 distinct for l<16
#define NSLAB (DIM / BK)   // 24

// One lane copies 16 contiguous bytes global -> LDS via the CDNA5 async engine
// (tracked by ASYNCcnt, no VGPR staging, no loadcnt serialization).
__device__ __forceinline__ void async_copy16(unsigned lds_addr, const float* gptr) {
    asm volatile("global_load_async_to_lds_b128 %0, %1, off"
                 :: "v"(lds_addr), "v"(gptr)
                 : "memory");
}

__device__ __forceinline__ void wait_async0() {
#if __has_builtin(__builtin_amdgcn_s_wait_asynccnt)
    __builtin_amdgcn_s_wait_asynccnt(0);
#else
    asm volatile("s_wait_asynccnt 0x0" ::: "memory");
#endif
}

// ---------------------------------------------------------------------------
// Kernel 1: per-batch GEMM dots = S·T^T via V_WMMA_F32_16X16X4_F32 with
// double-buffered async global->LDS staging; epilogue c = (1-cos(dots))/2
// -> d_out, plus deterministic per-block partial sums of (c, c^2).
// grid = (16, 16, 8), block = 256 (8 wave32)
// ---------------------------------------------------------------------------
__global__ __launch_bounds__(256) void ot_gemm_cos_kernel(
    const float* __restrict__ src, const float* __restrict__ tgt,
    float* __restrict__ out, float* __restrict__ psum, float* __restrict__ psq)
{
    __shared__ __align__(16) float As[2][TILE][LDSP];
    __shared__ __align__(16) float Bs[2][TILE][LDSP];
    __shared__ float red[512];

    const int b    = blockIdx.z;
    const int tm   = blockIdx.y * TILE;
    const int tn   = blockIdx.x * TILE;
    const int tid  = threadIdx.x;
    const int lane = tid & 31;
    const int wid  = tid >> 5;
    const int wr   = wid >> 1;      // wave row 0..3  -> m offset wr*16
    const int wc   = wid & 1;       // wave col 0..1  -> n offset wc*32 (two 16x16 tiles)
    const int hl   = lane >> 4;     // half-wave 0/1
    const int l16  = lane & 15;

    const float* A = src + (size_t)b * SEQ * DIM;
    const float* B = tgt + (size_t)b * SEQ * DIM;

    v8f acc0 = {};
    v8f acc1 = {};

    // async loader geometry: 64x32 floats = 512 16B-chunks per tile,
    // 2 chunks per thread per tile per slab.
    const int c0 = tid;            // chunk ids
    const int c1 = tid + 256;
    const int r0 = c0 >> 3, g0 = (c0 & 7) * 4;   // row 0..63, col (floats) 0..28
    const int r1 = c1 >> 3, g1 = (c1 & 7) * 4;

    const unsigned ldsA0 = (unsigned)(uintptr_t)&As[0][0][0];
    const unsigned ldsB0 = (unsigned)(uintptr_t)&Bs[0][0][0];
    const unsigned bufBytes = TILE * LDSP * 4;   // one buffer of one tile

    const unsigned offA0 = (unsigned)(r0 * LDSP + g0) * 4;
    const unsigned offA1 = (unsigned)(r1 * LDSP + g1) * 4;

    // ---- prologue: stage slab 0 into buffer 0 ----
    {
        const int kb = 0;
        async_copy16(ldsA0 + offA0, A + (size_t)(tm + r0) * DIM + kb + g0);
        async_copy16(ldsA0 + offA1, A + (size_t)(tm + r1) * DIM + kb + g1);
        async_copy16(ldsB0 + offA0, B + (size_t)(tn + r0) * DIM + kb + g0);
        async_copy16(ldsB0 + offA1, B + (size_t)(tn + r1) * DIM + kb + g1);
    }
    wait_async0();
    __syncthreads();

    for (int i = 0; i < NSLAB; ++i) {
        // ---- issue async copies for next slab into the other buffer ----
        if (i + 1 < NSLAB) {
            const int kb = (i + 1) * BK;
            const unsigned bo = ((i + 1) & 1) * bufBytes;
            async_copy16(ldsA0 + bo + offA0, A + (size_t)(tm + r0) * DIM + kb + g0);
            async_copy16(ldsA0 + bo + offA1, A + (size_t)(tm + r1) * DIM + kb + g1);
            async_copy16(ldsB0 + bo + offA0, B + (size_t)(tn + r0) * DIM + kb + g0);
            async_copy16(ldsB0 + bo + offA1, B + (size_t)(tn + r1) * DIM + kb + g1);
        }

        // ---- WMMA over the current buffer ----
        const float (*Ac)[LDSP] = As[i & 1];
        const float (*Bc)[LDSP] = Bs[i & 1];
        #pragma unroll
        for (int kk = 0; kk < BK; kk += 4) {
            const int k0 = kk + hl * 2;
            // A frag (16x4): lane = M (=l16), VGPR v + half -> K = hl*2 + v
            v2f a;
            a.x = Ac[wr * 16 + l16][k0];
            a.y = Ac[wr * 16 + l16][k0 + 1];
            // B frags (4x16): lane = N, VGPR v + half -> K = hl*2 + v
            v2f b0, b1;
            b0.x = Bc[wc * 32 + l16][k0];
            b0.y = Bc[wc * 32 + l16][k0 + 1];
            b1.x = Bc[wc * 32 + 16 + l16][k0];
            b1.y = Bc[wc * 32 + 16 + l16][k0 + 1];
            acc0 = __builtin_amdgcn_wmma_f32_16x16x4_f32(
                false, a, false, b0, (short)0, acc0, false, false);
            acc1 = __builtin_amdgcn_wmma_f32_16x16x4_f32(
                false, a, false, b1, (short)0, acc1, false, false);
        }

        // single barrier per slab: proves (a) next-slab async data landed,
        // (b) every wave finished reading the buffer that slab i+2 overwrites.
        wait_async0();
        __syncthreads();
    }

    // epilogue: c = (1 - cos(dots)) / 2 ; C/D layout: VGPR r -> M = r + hl*8, N = l16
    float lsum = 0.f, lsq = 0.f;
    const int gm  = tm + wr * 16 + hl * 8;
    const int gn0 = tn + wc * 32 + l16;
    #pragma unroll
    for (int r = 0; r < 8; ++r) {
        const float c0e = (1.0f - cosf(acc0[r])) * 0.5f;
        const float c1e = (1.0f - cosf(acc1[r])) * 0.5f;
        const size_t row = (size_t)b * SEQ * SEQ + (size_t)(gm + r) * SEQ;
        out[row + gn0]      = c0e;
        out[row + gn0 + 16] = c1e;
        lsum += c0e + c1e;
        lsq  += c0e * c0e + c1e * c1e;
    }

    red[tid] = lsum;
    red[256 + tid] = lsq;
    __syncthreads();
    for (int s = 128; s > 0; s >>= 1) {
        if (tid < s) { red[tid] += red[tid + s]; red[256 + tid] += red[256 + tid + s]; }
        __syncthreads();
    }
    if (tid == 0) {
        const int pid = blockIdx.z * 256 + blockIdx.y * 16 + blockIdx.x;
        psum[pid] = red[0];
        psq[pid]  = red[256];
    }
}

// ---------------------------------------------------------------------------
// Kernel 2: reduce 256 partials per batch -> 1/std (unbiased, ddof=1)
// grid = 8, block = 256
// ---------------------------------------------------------------------------
__global__ __launch_bounds__(256) void ot_std_kernel(
    const float* __restrict__ psum, const float* __restrict__ psq,
    float* __restrict__ invstd)
{
    __shared__ float s1[256];
    __shared__ float s2[256];
    const int b = blockIdx.x, t = threadIdx.x;
    s1[t] = psum[b * 256 + t];
    s2[t] = psq[b * 256 + t];
    __syncthreads();
    for (int s = 128; s > 0; s >>= 1) {
        if (t < s) { s1[t] += s1[t + s]; s2[t] += s2[t + s]; }
        __syncthreads();
    }
    if (t == 0) {
        const float N = 1048576.0f;  // 1024*1024
        const float sum = s1[0], sq = s2[0];
        const float var = (sq - sum * sum / N) / (N - 1.0f);
        invstd[b] = rsqrtf(fmaxf(var, 1e-30f));
    }
}

// ---------------------------------------------------------------------------
// Kernel 3: wave-per-row GEMVs: s_proj (rows 0..8191), t_proj (8192..16383),
// and row 16384 = dot(task_emb[task_id], W) + b_cost.
// grid = 2049, block = 256 (8 waves)
// ---------------------------------------------------------------------------
__global__ __launch_bounds__(256) void ot_proj_kernel(
    const float* __restrict__ src, const float* __restrict__ tgt,
    const float* __restrict__ W, const float* __restrict__ bc,
    const float* __restrict__ temb, const int* __restrict__ task_id,
    float* __restrict__ sproj, float* __restrict__ tproj, float* __restrict__ tpb)
{
    const int row  = blockIdx.x * 8 + (threadIdx.x >> 5);
    const int lane = threadIdx.x & 31;
    const float* base;
    if (row < 8192)        base = src  + (size_t)row * DIM;
    else if (row < 16384)  base = tgt  + (size_t)(row - 8192) * DIM;
    else if (row == 16384) base = temb + (size_t)task_id[0] * DIM;
    else return;

    float p = 0.f;
    for (int j = lane; j < DIM; j += 32) p += base[j] * W[j];
    #pragma unroll
    for (int off = 16; off > 0; off >>= 1) p += __shfl_down(p, off, 32);

    if (lane == 0) {
        if (row < 8192)       sproj[row] = p;
        else if (row < 16384) tproj[row - 8192] = p;
        else                  tpb[0] = p + bc[0];
    }
}

// ---------------------------------------------------------------------------
// Kernel 4: in-place finalize on d_out:
//   out = alpha * ( exp(-0.5*(c/std)^2) + sigmoid(s_proj + t_proj + taskbias) )
// grid = 8192, block = 256, 4 elements (float4) per thread
// ---------------------------------------------------------------------------
__global__ __launch_bounds__(256) void ot_finalize_kernel(
    float* __restrict__ out, const float* __restrict__ invstd,
    const float* __restrict__ sproj, const float* __restrict__ tproj,
    const float* __restrict__ tpb, const float* __restrict__ alpha_p)
{
    const size_t idx = ((size_t)blockIdx.x * 256 + threadIdx.x) * 4;
    const int b = (int)(idx >> 20);
    const int s = (int)((idx >> 10) & 1023);
    const int t = (int)(idx & 1023);

    const float inv   = invstd[b];
    const float cbase = sproj[b * SEQ + s] + tpb[0];
    const float alpha = alpha_p[0];

    const float4 c  = *(const float4*)(out + idx);
    const float4 tp = *(const float4*)(tproj + b * SEQ + t);

    float cv[4] = {c.x, c.y, c.z, c.w};
    float tv[4] = {tp.x, tp.y, tp.z, tp.w};
    float r[4];
    #pragma unroll
    for (int j = 0; j < 4; ++j) {
        const float cn = cv[j] * inv;
        const float g  = expf(-0.5f * cn * cn);
        const float cd = 1.0f / (1.0f + expf(-(cbase + tv[j])));
        r[j] = alpha * (g + cd);
    }
    *(float4*)(out + idx) = make_float4(r[0], r[1], r[2], r[3]);
}

// ---------------------------------------------------------------------------
// Inputs (setup_inputs order):
//  0 source_features [8,1024,768] f32   1 target_features [8,1024,768] f32
//  2 W_cost [768,1] f32   3 b_cost [1] f32   4 task_emb [4,768] f32
//  5 alpha [1] f32        6 task_id [1] int
// Output: [8,1024,1024] f32
// ---------------------------------------------------------------------------
extern "C" void kernel_launch(void* const* d_in, const int* in_sizes, int n_in,
                              void* d_out, int out_size, void* d_ws, size_t ws_size,
                              hipStream_t stream) {
    const float* src   = (const float*)d_in[0];
    const float* tgt   = (const float*)d_in[1];
    const float* W     = (const float*)d_in[2];
    const float* bc    = (const float*)d_in[3];
    const float* temb  = (const float*)d_in[4];
    const float* alpha = (const float*)d_in[5];
    const int*   tid   = (const int*)d_in[6];
    float* out = (float*)d_out;

    float* ws     = (float*)d_ws;
    float* psum   = ws;            // 2048
    float* psq    = ws + 2048;     // 2048
    float* invstd = ws + 4096;     // 8
    float* sproj  = ws + 4104;     // 8192
    float* tproj  = ws + 12296;    // 8192 (offset*4 bytes % 16 == 0 -> float4 ok)
    float* tpb    = ws + 20488;    // 1

    dim3 ggrid(SEQ / TILE, SEQ / TILE, 8);        // 16 x 16 x 8
    ot_gemm_cos_kernel<<<ggrid, 256, 0, stream>>>(src, tgt, out, psum, psq);
    ot_std_kernel<<<8, 256, 0, stream>>>(psum, psq, invstd);
    ot_proj_kernel<<<2049, 256, 0, stream>>>(src, tgt, W, bc, temb, tid,
                                             sproj, tproj, tpb);
    ot_finalize_kernel<<<8192, 256, 0, stream>>>(out, invstd, sproj, tproj,
                                                 tpb, alpha);
}